// Mamba2MCLMHeadModel_81157702025978
// MI455X (gfx1250) — compile-verified
//
#include <hip/hip_runtime.h>
#include <hip/hip_bf16.h>

// ---------------- model constants (match reference) ----------------
constexpr int V_    = 50277;
constexpr int D_    = 512;
constexpr int L_    = 2;
constexpr int DI_   = 1024;
constexpr int N_    = 64;
constexpr int H_    = 16;
constexpr int P_    = 64;
constexpr int DC_   = 4;
constexpr int DP_   = 64;
constexpr int SEG_  = 64;
constexpr int KTOP_ = 2;
constexpr int T_    = 256;
constexpr int SMAX_ = T_ / SEG_;                 // 4
constexpr int DPROJ_   = 2 * DI_ + 2 * N_ + H_;  // 2192
constexpr int CONVDIM_ = DI_ + 2 * N_;           // 1152
constexpr int HPN_  = H_ * P_ * N_;              // 65536
constexpr float EPS_ = 1e-5f;
constexpr float NEG_ = -1e30f;

// workspace layout (floats): per layer [conv ring | ssm | cst | cpl | psum], then hsn[T][D]
constexpr size_t WS_CONV = (size_t)CONVDIM_ * DC_;       // 4608
constexpr size_t WS_SSM  = (size_t)HPN_;                 // 65536
constexpr size_t WS_CST  = (size_t)SMAX_ * HPN_;         // 262144
constexpr size_t WS_CPL  = (size_t)SMAX_ * DP_;          // 256
constexpr size_t WS_PSUM = (size_t)DP_;                  // 64
constexpr size_t PER_L   = WS_CONV + WS_SSM + WS_CST + WS_CPL + WS_PSUM; // 332608
constexpr size_t WS_HSN  = 2 * PER_L;                    // offset of normalized hidden states

// ---------------- WMMA / async vector types (wave32, gfx1250) ----------------
typedef __attribute__((ext_vector_type(2))) float v2f;
typedef __attribute__((ext_vector_type(8))) float v8f;
typedef __attribute__((ext_vector_type(4))) int   v4i_;

// async global->LDS availability (toolchain-dependent; fall back to plain stores)
#if defined(__has_builtin)
#if __has_builtin(__builtin_amdgcn_global_load_async_to_lds_b128) && \
    __has_builtin(__builtin_amdgcn_s_wait_asynccnt)
#define HAS_ASYNC_LDS 1
#endif
#endif
#ifndef HAS_ASYNC_LDS
#define HAS_ASYNC_LDS 0
#endif

// ---------------- helpers ----------------
__device__ __forceinline__ float block_reduce_sum(float v, volatile float* s_red) {
    // wave32 intra-wave reduce
    #pragma unroll
    for (int o = 16; o > 0; o >>= 1) v += __shfl_xor(v, o, 32);
    const int wid = threadIdx.x >> 5;
    const int lid = threadIdx.x & 31;
    if (lid == 0) s_red[wid] = v;
    __syncthreads();
    if (wid == 0) {
        float r = s_red[lid];  // exactly 32 waves at blockDim==1024
        #pragma unroll
        for (int o = 16; o > 0; o >>= 1) r += __shfl_xor(r, o, 32);
        if (lid == 0) s_red[0] = r;
    }
    __syncthreads();
    float out = s_red[0];
    __syncthreads();
    return out;
}

__device__ __forceinline__ float sigmoidf_(float x) { return 1.0f / (1.0f + __expf(-x)); }
__device__ __forceinline__ float softplusf_(float x) {
    return (x > 20.0f) ? x : log1pf(__expf(x));
}

// ====================================================================
// Kernel 1: persistent sequential scan (single workgroup, 1024 threads)
// State lives in d_ws (L2-resident ~3MB); per-token intermediates in LDS.
// ====================================================================
__global__ __launch_bounds__(1024) void mamba_scan_kernel(
    const float* __restrict__ emb,    // [V, D]
    const float* __restrict__ normw,  // [L, D]
    const float* __restrict__ Win,    // [L, DPROJ, D]
    const float* __restrict__ Cw,     // [L, CONVDIM, DC]
    const float* __restrict__ Cb,     // [L, CONVDIM]
    const float* __restrict__ Alog,   // [L, H]
    const float* __restrict__ dtb,    // [L, H]
    const float* __restrict__ Dp,     // [L, H]
    const float* __restrict__ Gw,     // [L, DI]
    const float* __restrict__ Wout,   // [L, D, DI]
    const float* __restrict__ Wu,     // [L, DP, D]
    const float* __restrict__ ob,     // [L]
    const float* __restrict__ nf,     // [D]
    const int*   __restrict__ ids,    // [1, T]
    float* __restrict__ ws)
{
    const int tid = threadIdx.x;

    __shared__ __align__(16) float s_x[D_];
    __shared__ __align__(16) float s_u[D_];
    __shared__ __align__(16) float s_zx[DPROJ_];
    __shared__ __align__(16) float s_up[DP_];
    __shared__ __align__(16) float s_xbc[CONVDIM_];
    __shared__ __align__(16) float s_y[DI_];
    __shared__ __align__(16) float s_gv[DI_];
    __shared__ float s_dt[H_], s_dA[H_];
    __shared__ float s_gam[SMAX_ + 1];
    __shared__ float s_red[32];
    __shared__ int   s_cnt[L_], s_nc[L_];
    __shared__ int   s_flush, s_wi, s_newcnt;

    // zero-init persistent state (workspace is poisoned by the harness)
    for (size_t i = tid; i < 2 * PER_L; i += 1024) ws[i] = 0.0f;
    if (tid < L_) { s_cnt[tid] = 0; s_nc[tid] = 0; }
    float* hsn = ws + WS_HSN;
    __syncthreads();

    for (int t = 0; t < T_; ++t) {
        const int tok = ids[t];  // B==1
        // x = emb[tok]
        for (int i = tid; i < D_; i += 1024) s_x[i] = emb[(size_t)tok * D_ + i];
        __syncthreads();

        for (int l = 0; l < L_; ++l) {
            float* conv = ws + (size_t)l * PER_L;
            float* ssm  = conv + WS_CONV;
            float* cst  = ssm  + WS_SSM;
            float* cpl  = cst  + WS_CST;
            float* psum = cpl  + WS_CPL;

            // ---- 1. u = rms(x, norm_w[l]) ----
            float pv = 0.0f;
            for (int i = tid; i < D_; i += 1024) { float xv = s_x[i]; pv += xv * xv; }
            float ssq = block_reduce_sum(pv, s_red);
            float rs = rsqrtf(ssq / (float)D_ + EPS_);
            for (int i = tid; i < D_; i += 1024) s_u[i] = s_x[i] * rs * normw[l * D_ + i];
            __syncthreads();

            // ---- 2. GEMVs: zxbcdt = u @ Win.T ; u_proj = u @ Wu.T ----
            {
                const float* WinL = Win + (size_t)l * DPROJ_ * D_;
                const float* WuL  = Wu  + (size_t)l * DP_ * D_;
                const float4* u4  = (const float4*)s_u;
                for (int r = tid; r < DPROJ_ + DP_; r += 1024) {
                    const float* wrow = (r < DPROJ_) ? (WinL + (size_t)r * D_)
                                                     : (WuL + (size_t)(r - DPROJ_) * D_);
                    const float4* w4 = (const float4*)wrow;
                    float acc = 0.0f;
                    #pragma unroll 8
                    for (int d4 = 0; d4 < D_ / 4; ++d4) {
                        float4 w = w4[d4], uu = u4[d4];
                        acc += w.x * uu.x + w.y * uu.y + w.z * uu.z + w.w * uu.w;
                    }
                    if (r < DPROJ_) s_zx[r] = acc; else s_up[r - DPROJ_] = acc;
                }
            }
            __syncthreads();

            // ---- 3. conv ring update + depthwise conv + silu ----
            {
                const int ph_new = t & (DC_ - 1);
                for (int c = tid; c < CONVDIM_; c += 1024) {
                    float xin = s_zx[DI_ + c];
                    conv[c * DC_ + ph_new] = xin;
                    float acc = Cb[l * CONVDIM_ + c];
                    #pragma unroll
                    for (int j = 0; j < DC_; ++j) {
                        int phys = (t + 1 + j) & (DC_ - 1);
                        float tap = (phys == ph_new) ? xin : conv[c * DC_ + phys];
                        acc += tap * Cw[((size_t)l * CONVDIM_ + c) * DC_ + j];
                    }
                    s_xbc[c] = acc * sigmoidf_(acc);
                }
            }
            // ---- 4. dt = softplus(dt+dtb); dA = exp(dt * -exp(Alog)) ----
            if (tid < H_) {
                float dtv = softplusf_(s_zx[2 * DI_ + 2 * N_ + tid] + dtb[l * H_ + tid]);
                s_dt[tid] = dtv;
                s_dA[tid] = __expf(dtv * -__expf(Alog[l * H_ + tid]));
            }
            // ---- 5. pools / scores / top-k / softmax (tiny; one lane) ----
            if (tid == 0) {
                int cnt = s_cnt[l], nc = s_nc[l];
                float inv = 1.0f / (float)((cnt > 0) ? cnt : 1);
                float scores[SMAX_ + 1];
                for (int s = 0; s <= SMAX_; ++s) {
                    float acc = 0.0f;
                    for (int d = 0; d < DP_; ++d) {
                        float pool = (s < SMAX_) ? cpl[s * DP_ + d] : psum[d] * inv;
                        acc += s_up[d] * pool;
                    }
                    scores[s] = acc;
                }
                for (int s = 0; s < SMAX_; ++s) if (s >= nc) scores[s] = NEG_;
                scores[SMAX_] += ob[l];
                if (nc > KTOP_) {
                    int i1 = -1, i2 = -1; float b1 = -3.0e38f, b2 = -3.0e38f;
                    for (int s = 0; s < SMAX_; ++s) {
                        float vv = scores[s];
                        if (vv > b1) { b2 = b1; i2 = i1; b1 = vv; i1 = s; }
                        else if (vv > b2) { b2 = vv; i2 = s; }
                    }
                    for (int s = 0; s < SMAX_; ++s)
                        if (s != i1 && s != i2) scores[s] += NEG_;
                }
                float m = scores[0];
                for (int s = 1; s <= SMAX_; ++s) m = fmaxf(m, scores[s]);
                float den = 0.0f, e[SMAX_ + 1];
                for (int s = 0; s <= SMAX_; ++s) { e[s] = __expf(scores[s] - m); den += e[s]; }
                float rden = 1.0f / den;
                for (int s = 0; s <= SMAX_; ++s) s_gam[s] = e[s] * rden;
                int newcnt = cnt + 1;
                s_flush  = (newcnt >= SEG_) ? 1 : 0;
                s_wi     = (nc < SMAX_ - 1) ? nc : (SMAX_ - 1);
                s_newcnt = newcnt;
            }
            __syncthreads();

            // ---- 6. SSM update + fused mixture + y (one (h,p) per thread) ----
            {
                const int h = tid >> 6;
                const int p = tid & 63;
                const float dtv = s_dt[h], dAv = s_dA[h];
                const float xs  = s_xbc[h * P_ + p];
                const float dtx = dtv * xs;
                const float g0 = s_gam[0], g1 = s_gam[1], g2 = s_gam[2],
                            g3 = s_gam[3], g4 = s_gam[4];
                const int flush = s_flush;
                const int wi = s_wi;
                const size_t base = ((size_t)h * P_ + p) * N_;
                float4* ssm4 = (float4*)(ssm + base);
                const float4* c04 = (const float4*)(cst + 0 * (size_t)HPN_ + base);
                const float4* c14 = (const float4*)(cst + 1 * (size_t)HPN_ + base);
                const float4* c24 = (const float4*)(cst + 2 * (size_t)HPN_ + base);
                const float4* c34 = (const float4*)(cst + 3 * (size_t)HPN_ + base);
                float4* cw4 = (float4*)(cst + (size_t)wi * HPN_ + base);
                const float4* bm4 = (const float4*)(s_xbc + DI_);
                const float4* cc4 = (const float4*)(s_xbc + DI_ + N_);
                float yacc = 0.0f;
                #pragma unroll 4
                for (int q = 0; q < N_ / 4; ++q) {
                    float4 sv = ssm4[q];
                    float4 bm = bm4[q];
                    float4 cc = cc4[q];
                    float4 ns;
                    ns.x = sv.x * dAv + dtx * bm.x;
                    ns.y = sv.y * dAv + dtx * bm.y;
                    ns.z = sv.z * dAv + dtx * bm.z;
                    ns.w = sv.w * dAv + dtx * bm.w;
                    float4 a0 = c04[q], a1 = c14[q], a2 = c24[q], a3 = c34[q];
                    float fx = g4 * ns.x + g0 * a0.x + g1 * a1.x + g2 * a2.x + g3 * a3.x;
                    float fy = g4 * ns.y + g0 * a0.y + g1 * a1.y + g2 * a2.y + g3 * a3.y;
                    float fz = g4 * ns.z + g0 * a0.z + g1 * a1.z + g2 * a2.z + g3 * a3.z;
                    float fw = g4 * ns.w + g0 * a0.w + g1 * a1.w + g2 * a2.w + g3 * a3.w;
                    yacc += fx * cc.x + fy * cc.y + fz * cc.z + fw * cc.w;
                    ssm4[q] = ns;
                    if (flush) cw4[q] = ns;
                }
                s_y[tid] = yacc + Dp[l * H_ + h] * xs;
            }
            // ---- 6b. psum / cpl / counters ----
            if (tid < DP_) {
                float nsum = psum[tid] + s_up[tid];
                if (s_flush) {
                    cpl[s_wi * DP_ + tid] = nsum / (float)s_newcnt;
                    psum[tid] = 0.0f;
                } else {
                    psum[tid] = nsum;
                }
            }
            if (tid == 0) {
                if (s_flush) { s_cnt[l] = 0; s_nc[l] += 1; }
                else         { s_cnt[l] = s_newcnt; }
            }
            __syncthreads();

            // ---- 7. v = y*silu(z); gv = rms(v)*Gw ----
            float zv = s_zx[tid];
            float vv = s_y[tid] * (zv * sigmoidf_(zv));
            float ss2 = block_reduce_sum(vv * vv, s_red);
            float rs2 = rsqrtf(ss2 / (float)DI_ + EPS_);
            s_gv[tid] = vv * rs2 * Gw[l * DI_ + tid];
            __syncthreads();

            // ---- 8. x += gv @ Wout.T  (512 rows, 2 threads per row) ----
            {
                const float* WoutL = Wout + (size_t)l * D_ * DI_;
                const int r = tid >> 1, half = tid & 1;
                const float4* w4 = (const float4*)(WoutL + (size_t)r * DI_ + half * (DI_ / 2));
                const float4* g4 = (const float4*)(s_gv + half * (DI_ / 2));
                float acc = 0.0f;
                #pragma unroll 8
                for (int d4 = 0; d4 < DI_ / 8; ++d4) {
                    float4 w = w4[d4], g = g4[d4];
                    acc += w.x * g.x + w.y * g.y + w.z * g.z + w.w * g.w;
                }
                s_y[tid] = acc;  // reuse s_y as partial scratch
            }
            __syncthreads();
            if (tid < D_) s_x[tid] += s_y[2 * tid] + s_y[2 * tid + 1];
            __syncthreads();
        } // layers

        // ---- final RMSNorm for this token -> hsn[t] ----
        float pv3 = 0.0f;
        for (int i = tid; i < D_; i += 1024) { float xv = s_x[i]; pv3 += xv * xv; }
        float ss3 = block_reduce_sum(pv3, s_red);
        float rs3 = rsqrtf(ss3 / (float)D_ + EPS_);
        for (int i = tid; i < D_; i += 1024)
            hsn[(size_t)t * D_ + i] = s_x[i] * rs3 * nf[i];
        __syncthreads();
    } // tokens
}

// ====================================================================
// Kernel 2: logits GEMM out[t, v] = hsn[t,:] . emb[v,:]
// One block per 16-wide vocab strip; emb strip staged once into LDS
// (async global->LDS), 16 waves cover all 16 m-tiles so emb streams
// from HBM exactly once (~103 MB). WMMA f32 16x16x4, K=512.
// ====================================================================
constexpr int NTILES_N = (V_ + 15) / 16;           // 3143
constexpr int LROWP    = D_ + 4;                   // LDS row pitch (bank-conflict pad)

__global__ __launch_bounds__(512) void logits_wmma_kernel(
    const float* __restrict__ emb,   // [V, D]
    const float* __restrict__ hsn,   // [T, D]
    float* __restrict__ out)         // [1, T, V]
{
    __shared__ __align__(16) float s_emb[16 * LROWP];  // 16 vocab rows x 512 (padded)

    const int lane = threadIdx.x & 31;
    const int wave = threadIdx.x >> 5;               // 16 waves = 16 m-tiles
    const int n0   = blockIdx.x * 16;

    // ---- stage the 16 x 512 emb strip into LDS ----
    {
        // 2048 float4 chunks; 512 threads -> 4 chunks each; rows >= V clamp to V-1
        for (int c = threadIdx.x; c < 16 * (D_ / 4); c += 512) {
            const int row  = c >> 7;           // /128 chunks per row
            const int col4 = c & 127;
            const int vr   = n0 + row;
            const int vcl  = (vr < V_) ? vr : (V_ - 1);
            const float* gsrc = emb + (size_t)vcl * D_ + col4 * 4;
            float* ldst = s_emb + row * LROWP + col4 * 4;
#if HAS_ASYNC_LDS
            // param 1 per clang diagnostic: int4 in addrspace(1) (global), non-const
            __builtin_amdgcn_global_load_async_to_lds_b128(
                (__attribute__((address_space(1))) v4i_*)gsrc,
                (__attribute__((address_space(3))) v4i_*)ldst,
                0, 0);
#else
            *(float4*)ldst = *(const float4*)gsrc;
#endif
        }
#if HAS_ASYNC_LDS
        __builtin_amdgcn_s_wait_asynccnt(0);
#endif
        __syncthreads();
    }

    // ---- per-wave 16x16 tile: rows t0..t0+15, cols n0..n0+15 ----
    const int t0  = wave * 16;
    const int sub = lane & 15;
    const int khi = (lane >> 4) * 2;                 // K offset 0 or 2
    const int vrow = n0 + sub;

    // A layout (32-bit 16x4): lane L, VGPR j -> M = L%16, K = j + 2*(L/16)
    const float* arow = hsn + (size_t)(t0 + sub) * D_ + khi;
    // B layout symmetric: K = j + 2*(L/16), N = L%16 -> emb[n][k] from LDS
    const float* brow = s_emb + sub * LROWP + khi;

    v8f c = {0.f, 0.f, 0.f, 0.f, 0.f, 0.f, 0.f, 0.f};
    #pragma unroll 4
    for (int k = 0; k < D_; k += 4) {
        __builtin_prefetch(arow + k + 128, 0, 1);    // global_prefetch_b8 (A stream)
        float2 af = *(const float2*)(arow + k);      // global (L2-resident)
        float2 bf = *(const float2*)(brow + k);      // LDS (ds_load_b64)
        v2f a, b;
        a.x = af.x; a.y = af.y;
        b.x = bf.x; b.y = bf.y;
        c = __builtin_amdgcn_wmma_f32_16x16x4_f32(
                /*neg_a=*/false, a, /*neg_b=*/false, b,
                /*c_mod=*/(short)0, c, /*reuse_a=*/false, /*reuse_b=*/false);
    }

    // D layout: VGPR r, lanes 0-15 -> M=r, lanes 16-31 -> M=r+8; N = lane%16
    if (vrow < V_) {
        const int rbase = t0 + 8 * (lane >> 4);
        #pragma unroll
        for (int r = 0; r < 8; ++r) {
            out[(size_t)(rbase + r) * V_ + vrow] = c[r];
        }
    }
}

// ====================================================================
extern "C" void kernel_launch(void* const* d_in, const int* in_sizes, int n_in,
                              void* d_out, int out_size, void* d_ws, size_t ws_size,
                              hipStream_t stream) {
    const float* emb   = (const float*)d_in[0];   // embedding [V, D]
    const float* normw = (const float*)d_in[1];   // norm_w [L, D]
    const float* Win   = (const float*)d_in[2];   // in_proj_w [L, 2192, 512]
    const float* Cw    = (const float*)d_in[3];   // conv_w [L, 1152, 4]
    const float* Cb    = (const float*)d_in[4];   // conv_b [L, 1152]
    const float* Alog  = (const float*)d_in[5];   // A_log [L, 16]
    const float* dtb   = (const float*)d_in[6];   // dt_bias [L, 16]
    const float* Dp    = (const float*)d_in[7];   // D_param [L, 16]
    const float* Gw    = (const float*)d_in[8];   // gnorm_w [L, 1024]
    const float* Wout  = (const float*)d_in[9];   // out_proj_w [L, 512, 1024]
    const float* Wu    = (const float*)d_in[10];  // Wu_w [L, 64, 512]
    const float* ob    = (const float*)d_in[11];  // online_bias [L]
    const float* nf    = (const float*)d_in[12];  // norm_f_w [512]
    const int*   ids   = (const int*)d_in[13];    // input_ids [1, 256]
    float* ws  = (float*)d_ws;
    float* out = (float*)d_out;                   // [1, 256, 50277] fp32

    // Phase 1: sequential recurrence (persistent single workgroup; state in L2)
    mamba_scan_kernel<<<1, 1024, 0, stream>>>(
        emb, normw, Win, Cw, Cb, Alog, dtb, Dp, Gw, Wout, Wu, ob, nf, ids, ws);

    // Phase 2: logits GEMM; emb streamed once, B from LDS, fp32 WMMA pipe
    logits_wmma_kernel<<<NTILES_N, 512, 0, stream>>>(emb, ws + WS_HSN, out);
}